// LossFreeTags_83880711290947
// MI455X (gfx1250) — compile-verified
//
#include <hip/hip_runtime.h>
#include <hip/hip_bf16.h>

typedef float v2f __attribute__((ext_vector_type(2)));
typedef float v8f __attribute__((ext_vector_type(8)));

#define NB_DET 544   // 32 (b,s) * 17 channels: one block per det plane
#define HW 16384
#define HW4 4096

__device__ static inline float waveSum(float v) {
#pragma unroll
  for (int off = 16; off > 0; off >>= 1) v += __shfl_xor(v, off, 32);
  return v;
}

// ---------------------------------------------------------------------------
// Kernel 1: det term partial sums.  One block per (bs, channel) plane of
// 4096 float4s; 16 uniform iterations of 256 threads -> no tail, no
// divergence, EXEC all-ones at every WMMA.  Accumulation runs through the
// matrix pipe: V_WMMA_F32_16X16X4_F32 with an all-ones B operand folds all
// 128 per-wave error values into the striped C accumulator each iteration (two issues).
// ---------------------------------------------------------------------------
__global__ __launch_bounds__(256) void det_loss_kernel(
    const float4* __restrict__ preds, const float4* __restrict__ masks,
    const float4* __restrict__ heat, float* __restrict__ partials) {
  const int bs  = blockIdx.x / 17;   // scalar, once per block
  const int cch = blockIdx.x % 17;
  const int b   = bs >> 2;           // / S

  const float4* __restrict__ pPred = preds + ((long)bs * 34 + cch) * HW4;
  const float4* __restrict__ pHeat = heat + ((long)b * 17 + cch) * HW4;
  const float4* __restrict__ pMask = masks + (long)b * HW4;

  v8f c = {};              // 16x16 f32 accumulator, striped across the wave
  v2f onesB = {1.f, 1.f};  // B = 4x16 all-ones

#pragma unroll 2
  for (int it = 0; it < 16; ++it) {
    int hw = threadIdx.x + (it << 8);
    float4 d  = pPred[hw];
    float4 h  = pHeat[hw];
    float4 mk = pMask[hw];
    float dx = d.x - h.x, dy = d.y - h.y, dz = d.z - h.z, dw = d.w - h.w;
    v2f a0 = {dx * dx * mk.x, dy * dy * mk.y};
    v2f a1 = {dz * dz * mk.z, dw * dw * mk.w};
    c = __builtin_amdgcn_wmma_f32_16x16x4_f32(false, a0, false, onesB,
                                              (short)0, c, false, false);
    c = __builtin_amdgcn_wmma_f32_16x16x4_f32(false, a1, false, onesB,
                                              (short)0, c, false, false);
  }

  // Every column n of C carries an identical total; column n lives in lanes
  // n (rows 0..7) and n+16 (rows 8..15).  Extract column 0.
  float p = c[0] + c[1] + c[2] + c[3] + c[4] + c[5] + c[6] + c[7];
  float wsum = __shfl(p, 0, 32) + __shfl(p, 16, 32);

  __shared__ float sw[8];
  int lane = threadIdx.x & 31, wid = threadIdx.x >> 5;
  if (lane == 0) sw[wid] = wsum;
  __syncthreads();
  if (threadIdx.x == 0) {
    float s = 0.f;
    for (int w = 0; w < 8; ++w) s += sw[w];
    partials[blockIdx.x] = s;
  }
}

// ---------------------------------------------------------------------------
// Kernel 2: pull + push tag loss.  One wave per batch image b; lane p owns
// person p (<30).  Writes out[b] = sum_s 0.001*(push[b,s] + pull[b,s]).
// ---------------------------------------------------------------------------
__global__ __launch_bounds__(32) void tag_loss_kernel(
    const float* __restrict__ preds, const int* __restrict__ kp,
    float* __restrict__ out) {
  const int b = blockIdx.x;
  const int lane = threadIdx.x;
  const float EPSf = 1e-6f;
  __shared__ float sMean[4][30];
  __shared__ float sValid[30];

  int idxs[17];
  float visf[17];
  float cnt = 0.f;
  if (lane < 30) {
#pragma unroll
    for (int k = 0; k < 17; ++k) {
      int base = ((b * 30 + lane) * 17 + k) * 2;
      idxs[k] = kp[base];
      float v = (kp[base + 1] > 0) ? 1.f : 0.f;
      visf[k] = v;
      cnt += v;
    }
  } else {
#pragma unroll
    for (int k = 0; k < 17; ++k) { idxs[k] = 0; visf[k] = 0.f; }
  }
  float validF = (cnt > 0.f) ? 1.f : 0.f;
  if (lane < 30) sValid[lane] = validF;
  float safe = fmaxf(cnt, 1.f);
  float num = waveSum(validF);

  float myMean[4];
  float pullv[4];
#pragma unroll
  for (int s = 0; s < 4; ++s) {
    float g[17];
    float m = 0.f;
    if (lane < 30) {
#pragma unroll
      for (int k = 0; k < 17; ++k) {
        int id = idxs[k];
        int ch = id >> 14;        // / HW
        int pix = id & (HW - 1);
        g[k] = preds[((long)(b * 4 + s) * 34 + 17 + ch) * HW + pix];
        m += g[k] * visf[k];
      }
    } else {
#pragma unroll
      for (int k = 0; k < 17; ++k) g[k] = 0.f;
    }
    m /= safe;
    myMean[s] = m;
    float pp = 0.f;
#pragma unroll
    for (int k = 0; k < 17; ++k) {
      float d = g[k] - m;
      pp += d * d * visf[k];
    }
    pp /= safe;
    if (lane < 30) sMean[s][lane] = m;
    pullv[s] = waveSum(pp * validF) / (num + EPSf);
  }
  __syncthreads();

  float pushAcc[4] = {0.f, 0.f, 0.f, 0.f};
  if (lane < 30) {
    for (int j = 0; j < 30; ++j) {
      float pm = validF * sValid[j];
#pragma unroll
      for (int s = 0; s < 4; ++s) {
        float d = myMean[s] - sMean[s][j];
        pushAcc[s] += expf(-d * d) * pm;
      }
    }
  }
  float total = 0.f;
  float denom = (num - 1.f) * num + EPSf;
#pragma unroll
  for (int s = 0; s < 4; ++s) {
    float ps = (waveSum(pushAcc[s]) - num) / denom * 0.5f;
    total += 0.001f * (ps + pullv[s]);
  }
  if (lane == 0) out[b] = total;
}

// ---------------------------------------------------------------------------
// Kernel 3: combine det partials (NB_DET) + tag partials (8) into the scalar.
// ---------------------------------------------------------------------------
__global__ __launch_bounds__(256) void finalize_kernel(
    const float* __restrict__ ws, float* __restrict__ out) {
  __shared__ float red[256];
  const float detScale = 1.0f / (32.0f * 278528.0f);  // /(B*S) /(17*H*W)
  const float tagScale = 1.0f / 32.0f;                // /(B*S)
  float a = 0.f;
  for (int i = threadIdx.x; i < NB_DET; i += 256) a += ws[i] * detScale;
  if (threadIdx.x < 8) a += ws[NB_DET + threadIdx.x] * tagScale;
  red[threadIdx.x] = a;
  __syncthreads();
  for (int off = 128; off > 0; off >>= 1) {
    if (threadIdx.x < off) red[threadIdx.x] += red[threadIdx.x + off];
    __syncthreads();
  }
  if (threadIdx.x == 0) out[0] = red[0];
}

extern "C" void kernel_launch(void* const* d_in, const int* in_sizes, int n_in,
                              void* d_out, int out_size, void* d_ws,
                              size_t ws_size, hipStream_t stream) {
  const float* preds = (const float*)d_in[0];    // (8,4,34,128,128) f32
  const float* masks = (const float*)d_in[1];    // (8,128,128) f32
  const float* heat  = (const float*)d_in[2];    // (8,17,128,128) f32
  const int*   kp    = (const int*)d_in[3];      // (8,30,17,2) int
  float* out = (float*)d_out;                    // scalar
  float* ws  = (float*)d_ws;                     // NB_DET + 8 floats

  det_loss_kernel<<<NB_DET, 256, 0, stream>>>(
      (const float4*)preds, (const float4*)masks, (const float4*)heat, ws);
  tag_loss_kernel<<<8, 32, 0, stream>>>(preds, kp, ws + NB_DET);
  finalize_kernel<<<1, 256, 0, stream>>>(ws, out);
}